// CTCLayer_75110388072637
// MI455X (gfx1250) — compile-verified
//
#include <hip/hip_runtime.h>

// CTC batch cost (Keras semantics): B=128, T=1024, C=128 (blank=C-1), L=128, S=2L+1=257.
// One workgroup per batch element; thread s owns extended-label state s.
// Alpha recursion double-buffered in LDS; emission rows (512B each) staged one
// frame ahead via CDNA5 async global->LDS loads (ASYNCcnt / s_wait_asynccnt).
// Scan terminates exactly at input_len (clamped to T; later frames are never run).

#define B_  128
#define T_  1024
#define C_  128
#define L_  128
#define S_  257
#define NEGV (-1e30f)
#define EPS_ 1e-7f

__device__ __forceinline__ float lae(float a, float b) {
    float m = fmaxf(a, b);
    float d = fabsf(a - b);
    return m + __logf(1.0f + __expf(-d));  // -1e30 surrogate: exp underflows to 0
}

__global__ __launch_bounds__(288) void ctc_alpha_kernel(
    const int*   __restrict__ y_true,     // [B,L]
    const float* __restrict__ y_pred,     // [B,T,C] softmax probs
    const int*   __restrict__ input_len,  // [B]
    const int*   __restrict__ label_len,  // [B]
    float*       __restrict__ out)        // [B]
{
    __shared__ float raw[2][C_];          // staged emission rows (async target)
    __shared__ float alphaBuf[2][S_ + 2]; // [k][0..1] = NEG pads, state s at [k][2+s]

    const int b   = blockIdx.x;
    const int tid = threadIdx.x;
    const float* base = y_pred + (size_t)b * T_ * C_;
    int steps = input_len[b];
    if (steps > T_) steps = T_;           // uniform per block: exact early stop

    // ---- init alpha + static pads ----
    if (tid < 2) { alphaBuf[0][tid] = NEGV; alphaBuf[1][tid] = NEGV; }
    if (tid < S_) alphaBuf[0][2 + tid] = (tid == 0) ? 0.0f : NEGV;

    // ---- loop-invariant per-state data: extended symbol + skip permission ----
    int  ext_s = C_ - 1;   // blank
    bool allow = false;
    if (tid < S_ && (tid & 1)) {
        int i = (tid - 1) >> 1;
        ext_s = y_true[b * L_ + i];
        if (tid >= 3) allow = (ext_s != y_true[b * L_ + i - 1]);
    }

    // ---- prefetch frame 0: wave 0, 16B per lane = 512B row ----
    if (tid < 32) {
        unsigned lds  = (unsigned)(unsigned long long)&raw[0][tid * 4];
        unsigned voff = (unsigned)(tid * 16);
        asm volatile("global_load_async_to_lds_b128 %0, %1, %2"
                     :: "v"(lds), "v"(voff), "s"(base) : "memory");
    }
    asm volatile("s_wait_asynccnt 0" ::: "memory");
    __syncthreads();

#define CTC_STEP(CUR, NXT, TCUR)                                              \
    {                                                                         \
        if (tid < 32 && ((TCUR) + 1) < steps) {                               \
            unsigned lds  = (unsigned)(unsigned long long)&raw[NXT][tid * 4]; \
            unsigned voff = (unsigned)((((TCUR) + 1) * C_ + tid * 4) * 4);    \
            asm volatile("global_load_async_to_lds_b128 %0, %1, %2"           \
                         :: "v"(lds), "v"(voff), "s"(base) : "memory");       \
        }                                                                     \
        if (tid < S_) {                                                       \
            float lp = __logf(raw[CUR][ext_s] + EPS_);                        \
            float a0 = alphaBuf[CUR][2 + tid];                                \
            float a1 = alphaBuf[CUR][1 + tid];                                \
            float a2 = allow ? alphaBuf[CUR][tid] : NEGV;                     \
            alphaBuf[NXT][2 + tid] = lp + lae(lae(a0, a1), a2);               \
        }                                                                     \
        asm volatile("s_wait_asynccnt 0" ::: "memory");                       \
        __syncthreads();                                                      \
    }

    // unrolled-by-2 scan; pair iterations preserve "current buffer == 0"
    int t = 0;
    while (t + 1 < steps) {
        CTC_STEP(0, 1, t)
        CTC_STEP(1, 0, t + 1)
        t += 2;
    }
    int cur = 0;
    if (t < steps) {          // odd tail
        CTC_STEP(0, 1, t)
        cur = 1;
    }
#undef CTC_STEP

    if (tid == 0) {
        int lab = label_len[b];
        int ilast = 2 * lab;
        int iprev = (ilast - 1) > 0 ? (ilast - 1) : 0;
        float alast = alphaBuf[cur][2 + ilast];
        float aprev = alphaBuf[cur][2 + iprev];
        out[b] = -lae(alast, aprev);
    }
}

extern "C" void kernel_launch(void* const* d_in, const int* in_sizes, int n_in,
                              void* d_out, int out_size, void* d_ws, size_t ws_size,
                              hipStream_t stream) {
    const int*   y_true    = (const int*)d_in[0];
    const float* y_pred    = (const float*)d_in[1];
    const int*   input_len = (const int*)d_in[2];
    const int*   label_len = (const int*)d_in[3];
    float*       out       = (float*)d_out;
    (void)in_sizes; (void)n_in; (void)out_size; (void)d_ws; (void)ws_size;

    ctc_alpha_kernel<<<B_, 288, 0, stream>>>(y_true, y_pred, input_len, label_len, out);
}